// ModelClass_27779848471455
// MI455X (gfx1250) — compile-verified
//
#include <hip/hip_runtime.h>

#define N_NODES 100000
#define N_EDGES 1600000
#define DIM 64
#define N_GRAPHS 512
#define N_CONVS 4
#define DJ (DIM * (N_CONVS + 1))   // 320
#define BN_EPS 1e-5f

typedef __attribute__((ext_vector_type(2))) float v2f;
typedef __attribute__((ext_vector_type(8))) float v8f;

// ---------------------------------------------------------------------------
// Wave-level fp32 WMMA tile: acc += A[16xK] * B[Kx16] via chained 16x16x4.
// A: row-major, lda = row stride, pointer at tile row 0.
// BT: TRANSPOSED B (BT[n][k]), ldbt = row stride, pointer at tile col 0
//     (i.e. BT + col0*ldbt). Transposing makes the per-lane B fetch a
//     contiguous b64, matching the A-side access pattern.
// ISA layouts: A lane m=lane%16, K = k + 2*(lane/16) + {0,1} across 2 VGPRs.
//              B lane n=lane%16, same K split. C/D: row = j + 8*(lane/16).
// ---------------------------------------------------------------------------
__device__ __forceinline__ v8f wmma_acc_f32(const float* __restrict__ A, int lda,
                                            const float* __restrict__ BT, int ldbt,
                                            int K, v8f acc) {
  const int lane = threadIdx.x & 31;
  const int half = lane >> 4;
  const int mn   = lane & 15;
  for (int k = 0; k < K; k += 4) {
    const int ka = k + 2 * half;
    v2f a, b;
    a.x = A[mn * lda + ka];
    a.y = A[mn * lda + ka + 1];
    b.x = BT[mn * ldbt + ka];
    b.y = BT[mn * ldbt + ka + 1];
    acc = __builtin_amdgcn_wmma_f32_16x16x4_f32(false, a, false, b,
                                                (short)0, acc, false, false);
  }
  return acc;
}

// WT[n*K + k] = W[k*N + n]
__global__ void transpose_kernel(const float* __restrict__ W, float* __restrict__ WT,
                                 int K, int N) {
  int id = blockIdx.x * blockDim.x + threadIdx.x;
  if (id >= K * N) return;
  int k = id / N;
  int n = id - k * N;
  WT[(long)n * K + k] = W[id];
}

// ---------------------------------------------------------------------------
// Fused pre-MLP: h_raw = PReLU(PReLU(x@W1+b1)@W2+b2); accumulate BN stats.
// Block = 128 threads (4 waves); block handles 16 rows x 64 cols.
// ---------------------------------------------------------------------------
__global__ void pre_mlp_kernel(const float* __restrict__ x,
                               const float* __restrict__ W1T, const float* __restrict__ b1,
                               const float* __restrict__ a1,
                               const float* __restrict__ W2T, const float* __restrict__ b2,
                               const float* __restrict__ a2,
                               float* __restrict__ h, float* __restrict__ sums) {
  __shared__ float t[16][DIM + 4];
  const int wave = threadIdx.x >> 5;
  const int lane = threadIdx.x & 31;
  const int half = lane >> 4;
  const int n    = lane & 15;
  const int row0 = blockIdx.x * 16;
  const int nb   = wave * 16;

  v8f acc = {};
  acc = wmma_acc_f32(x + (long)row0 * DIM, DIM, W1T + (long)nb * DIM, DIM, DIM, acc);
  const float bb1 = b1[nb + n];
  const float al1 = a1[nb + n];
#pragma unroll
  for (int j = 0; j < 8; ++j) {
    float v = acc[j] + bb1;
    v = (v >= 0.f) ? v : al1 * v;
    t[j + 8 * half][nb + n] = v;
  }
  __syncthreads();

  v8f acc2 = {};
  acc2 = wmma_acc_f32(&t[0][0], DIM + 4, W2T + (long)nb * DIM, DIM, DIM, acc2);
  const float bb2 = b2[nb + n];
  const float al2 = a2[nb + n];
  float s = 0.f, sq = 0.f;
#pragma unroll
  for (int j = 0; j < 8; ++j) {
    float v = acc2[j] + bb2;
    v = (v >= 0.f) ? v : al2 * v;
    h[(long)(row0 + j + 8 * half) * DIM + nb + n] = v;
    s += v;
    sq += v * v;
  }
  atomicAdd(&sums[nb + n], s);
  atomicAdd(&sums[DIM + nb + n], sq);
}

// out = A[rows,64] @ W[64,64] + bias  (per-node message transform, WT input)
__global__ void node_gemm64_kernel(const float* __restrict__ A,
                                   const float* __restrict__ WT,
                                   const float* __restrict__ bias,
                                   float* __restrict__ out) {
  const int wave = threadIdx.x >> 5;
  const int lane = threadIdx.x & 31;
  const int half = lane >> 4;
  const int n    = lane & 15;
  const int row0 = blockIdx.x * 16;
  const int nb   = wave * 16;
  v8f acc = {};
  acc = wmma_acc_f32(A + (long)row0 * DIM, DIM, WT + (long)nb * DIM, DIM, DIM, acc);
  const float b = bias[nb + n];
#pragma unroll
  for (int j = 0; j < 8; ++j)
    out[(long)(row0 + j + 8 * half) * DIM + nb + n] = acc[j] + b;
}

__global__ void bn_prep_kernel(const float* __restrict__ sums,
                               const float* __restrict__ gamma,
                               const float* __restrict__ beta,
                               float* __restrict__ scale, float* __restrict__ shift) {
  int c = threadIdx.x;
  if (c < DIM) {
    float mu  = sums[c] * (1.f / (float)N_NODES);
    float var = sums[DIM + c] * (1.f / (float)N_NODES) - mu * mu;
    float sc  = gamma[c] / sqrtf(var + BN_EPS);
    scale[c]  = sc;
    shift[c]  = beta[c] - mu * sc;
  }
}

__global__ void bn_apply_kernel(float* __restrict__ h,
                                const float* __restrict__ scale,
                                const float* __restrict__ shift) {
  int idx = blockIdx.x * blockDim.x + threadIdx.x;      // over N*16 float4 chunks
  if (idx >= N_NODES * 16) return;
  int c = (idx & 15) * 4;
  float4 v = ((float4*)h)[idx];
  v.x = v.x * scale[c + 0] + shift[c + 0];
  v.y = v.y * scale[c + 1] + shift[c + 1];
  v.z = v.z * scale[c + 2] + shift[c + 2];
  v.w = v.w * scale[c + 3] + shift[c + 3];
  ((float4*)h)[idx] = v;
}

// aggr[dst] += m[src] ; one thread per (edge, 4-col chunk)
__global__ void edge_scatter_kernel(const float* __restrict__ m,
                                    const int* __restrict__ src,
                                    const int* __restrict__ dst,
                                    float* __restrict__ aggr) {
  int idx = blockIdx.x * blockDim.x + threadIdx.x;      // E*16 chunks
  int e = idx >> 4;
  int c = (idx & 15) * 4;
  int s = src[e];
  int d = dst[e];
  float4 v = *(const float4*)(m + (long)s * DIM + c);
  float* p = aggr + (long)d * DIM + c;
  atomicAdd(p + 0, v.x);
  atomicAdd(p + 1, v.y);
  atomicAdd(p + 2, v.z);
  atomicAdd(p + 3, v.w);
}

// h = prelu(aggr)   (aggr already contains self term + messages)
__global__ void prelu_kernel(const float* __restrict__ aggr,
                             const float* __restrict__ alpha,
                             float* __restrict__ h) {
  int idx = blockIdx.x * blockDim.x + threadIdx.x;      // N*16 chunks
  if (idx >= N_NODES * 16) return;
  int c = (idx & 15) * 4;
  float4 v = ((const float4*)aggr)[idx];
  v.x = (v.x >= 0.f) ? v.x : alpha[c + 0] * v.x;
  v.y = (v.y >= 0.f) ? v.y : alpha[c + 1] * v.y;
  v.z = (v.z >= 0.f) ? v.z : alpha[c + 2] * v.z;
  v.w = (v.w >= 0.f) ? v.w : alpha[c + 3] * v.w;
  ((float4*)h)[idx] = v;
}

// global_add_pool of one JK stage into g[:, colbase:colbase+64].
// batchidx is sorted, so run-length accumulate RUN consecutive nodes in
// registers and flush once per run -> ~RUNx fewer contended L2 atomics.
#define POOL_RUN 8
__global__ void pool_kernel(const float* __restrict__ h,
                            const int* __restrict__ batchidx,
                            float* __restrict__ g, int colbase) {
  int id = blockIdx.x * blockDim.x + threadIdx.x;       // (N/RUN)*16 threads
  if (id >= (N_NODES / POOL_RUN) * 16) return;
  int chunk = (id & 15) * 4;
  int n0 = (id >> 4) * POOL_RUN;

  int cur = batchidx[n0];
  float4 accv = *(const float4*)(h + (long)n0 * DIM + chunk);
#pragma unroll
  for (int k = 1; k < POOL_RUN; ++k) {
    int node = n0 + k;
    int b = batchidx[node];
    float4 v = *(const float4*)(h + (long)node * DIM + chunk);
    if (b != cur) {
      float* p = g + (long)cur * DJ + colbase + chunk;
      atomicAdd(p + 0, accv.x);
      atomicAdd(p + 1, accv.y);
      atomicAdd(p + 2, accv.z);
      atomicAdd(p + 3, accv.w);
      cur = b;
      accv = v;
    } else {
      accv.x += v.x; accv.y += v.y; accv.z += v.z; accv.w += v.w;
    }
  }
  float* p = g + (long)cur * DJ + colbase + chunk;
  atomicAdd(p + 0, accv.x);
  atomicAdd(p + 1, accv.y);
  atomicAdd(p + 2, accv.z);
  atomicAdd(p + 3, accv.w);
}

// act = leaky_relu(g[512,320] @ W[320,320] + b)   (WT input)
__global__ void ffn1_kernel(const float* __restrict__ g,
                            const float* __restrict__ WT,
                            const float* __restrict__ bias,
                            float* __restrict__ act) {
  const int wave = threadIdx.x >> 5;
  const int lane = threadIdx.x & 31;
  const int half = lane >> 4;
  const int n    = lane & 15;
  const int row0 = blockIdx.x * 16;
  const int nb   = blockIdx.y * 64 + wave * 16;
  v8f acc = {};
  acc = wmma_acc_f32(g + (long)row0 * DJ, DJ, WT + (long)nb * DJ, DJ, DJ, acc);
  const float b = bias[nb + n];
#pragma unroll
  for (int j = 0; j < 8; ++j) {
    float v = acc[j] + b;
    v = (v >= 0.f) ? v : 0.01f * v;
    act[(long)(row0 + j + 8 * half) * DJ + nb + n] = v;
  }
}

// out[g] = act[g,:] . W2[:,0] + b2
__global__ void ffn2_kernel(const float* __restrict__ act,
                            const float* __restrict__ W2,
                            const float* __restrict__ b2,
                            float* __restrict__ out) {
  int row = blockIdx.x;
  int lane = threadIdx.x;
  float s = 0.f;
  for (int k = lane; k < DJ; k += 32) s += act[(long)row * DJ + k] * W2[k];
#pragma unroll
  for (int off = 16; off > 0; off >>= 1) s += __shfl_down(s, off, 32);
  if (lane == 0) out[row] = s + b2[0];
}

__global__ void zero_kernel(float* __restrict__ p, int n) {
  int i = blockIdx.x * blockDim.x + threadIdx.x;
  if (i < n) p[i] = 0.f;
}

extern "C" void kernel_launch(void* const* d_in, const int* in_sizes, int n_in,
                              void* d_out, int out_size, void* d_ws, size_t ws_size,
                              hipStream_t stream) {
  const float* x        = (const float*)d_in[0];
  const int*   ei       = (const int*)d_in[1];
  const int*   batchidx = (const int*)d_in[2];
  const float* W_pre1   = (const float*)d_in[3];
  const float* b_pre1   = (const float*)d_in[4];
  const float* a_pre1   = (const float*)d_in[5];
  const float* W_pre2   = (const float*)d_in[6];
  const float* b_pre2   = (const float*)d_in[7];
  const float* a_pre2   = (const float*)d_in[8];
  const float* bn_g     = (const float*)d_in[9];
  const float* bn_b     = (const float*)d_in[10];
  const float* W_msg    = (const float*)d_in[11];   // [4,64,64]
  const float* b_msg    = (const float*)d_in[12];   // [4,64]
  const float* a_act    = (const float*)d_in[13];
  const float* W_f1     = (const float*)d_in[14];   // [320,320]
  const float* b_f1     = (const float*)d_in[15];
  const float* W_f2     = (const float*)d_in[16];   // [320,1]
  const float* b_f2     = (const float*)d_in[17];

  const int* src = ei;
  const int* dst = ei + N_EDGES;

  // workspace layout (floats)
  float* ws    = (float*)d_ws;
  float* h     = ws;                 ws += (size_t)N_NODES * DIM;
  float* m     = ws;                 ws += (size_t)N_NODES * DIM;
  float* aggr  = ws;                 ws += (size_t)N_NODES * DIM;
  float* g     = ws;                 ws += (size_t)N_GRAPHS * DJ;
  float* act   = ws;                 ws += (size_t)N_GRAPHS * DJ;
  float* sums  = ws;                 ws += 2 * DIM;
  float* scale = ws;                 ws += DIM;
  float* shift = ws;                 ws += DIM;
  float* W1T   = ws;                 ws += DIM * DIM;
  float* W2T   = ws;                 ws += DIM * DIM;
  float* WmT   = ws;                 ws += (size_t)N_CONVS * DIM * DIM;
  float* Wf1T  = ws;                 ws += (size_t)DJ * DJ;

  const int elemChunks = N_NODES * 16;                  // float4 chunks of [N,64]

  // one-time weight transposes (contiguous b64 B-operand fetches for WMMA)
  transpose_kernel<<<(DIM * DIM + 255) / 256, 256, 0, stream>>>(W_pre1, W1T, DIM, DIM);
  transpose_kernel<<<(DIM * DIM + 255) / 256, 256, 0, stream>>>(W_pre2, W2T, DIM, DIM);
  for (int i = 0; i < N_CONVS; ++i)
    transpose_kernel<<<(DIM * DIM + 255) / 256, 256, 0, stream>>>(
        W_msg + (size_t)i * DIM * DIM, WmT + (size_t)i * DIM * DIM, DIM, DIM);
  transpose_kernel<<<(DJ * DJ + 255) / 256, 256, 0, stream>>>(W_f1, Wf1T, DJ, DJ);

  // init accumulators (ws is poisoned; must be deterministic each call)
  zero_kernel<<<(N_GRAPHS * DJ + 255) / 256, 256, 0, stream>>>(g, N_GRAPHS * DJ);
  zero_kernel<<<1, 256, 0, stream>>>(sums, 2 * DIM);

  // pre-MLP (fused 2x linear + PReLU) with BN stat accumulation
  pre_mlp_kernel<<<N_NODES / 16, 128, 0, stream>>>(x, W1T, b_pre1, a_pre1,
                                                   W2T, b_pre2, a_pre2, h, sums);
  bn_prep_kernel<<<1, 64, 0, stream>>>(sums, bn_g, bn_b, scale, shift);
  bn_apply_kernel<<<elemChunks / 256, 256, 0, stream>>>(h, scale, shift);

  const int poolThreads = (N_NODES / POOL_RUN) * 16;
  const int poolGrid = (poolThreads + 255) / 256;

  // JK stage 0
  pool_kernel<<<poolGrid, 256, 0, stream>>>(h, batchidx, g, 0);

  for (int i = 0; i < N_CONVS; ++i) {
    // per-node message transform (replaces per-edge matmul): m = h@W_i + b_i
    node_gemm64_kernel<<<N_NODES / 16, 128, 0, stream>>>(
        h, WmT + (size_t)i * DIM * DIM, b_msg + (size_t)i * DIM, m);
    // self term: aggr = h
    hipMemcpyAsync(aggr, h, (size_t)N_NODES * DIM * sizeof(float),
                   hipMemcpyDeviceToDevice, stream);
    // aggr[dst] += m[src]
    edge_scatter_kernel<<<(N_EDGES * 16) / 256, 256, 0, stream>>>(m, src, dst, aggr);
    // h = prelu(aggr)
    prelu_kernel<<<elemChunks / 256, 256, 0, stream>>>(aggr, a_act, h);
    // JK stage i+1
    pool_kernel<<<poolGrid, 256, 0, stream>>>(h, batchidx, g, (i + 1) * DIM);
  }

  // FFN head
  ffn1_kernel<<<dim3(N_GRAPHS / 16, DJ / 64), 128, 0, stream>>>(g, Wf1T, b_f1, act);
  ffn2_kernel<<<N_GRAPHS, 32, 0, stream>>>(act, W_f2, b_f2, (float*)d_out);
}